// BoundingBoxDiscipline_29429115912855
// MI455X (gfx1250) — compile-verified
//
#include <hip/hip_runtime.h>
#include <climits>

#ifndef __has_builtin
#define __has_builtin(x) 0
#endif

// ---- tiling ---------------------------------------------------------------
#define CH 21
#define BLOCK 128                      // 4 waves (wave32)
#define TILE_PIX 512                   // pixels per tile; 2^18 % 512 == 0 -> tile never crosses a batch
#define TILE_BYTES (TILE_PIX * CH * 4) // 43008 B = 21 rounds * 128 lanes * 16 B
#define COPY_ROUNDS (TILE_BYTES / (BLOCK * 16))
#define PIX_PER_THREAD (TILE_PIX / BLOCK)

// ---- CDNA5 async global->LDS copy (ASYNCcnt path) -------------------------
typedef int v4i_t __attribute__((vector_size(16)));

__device__ __forceinline__ void async_copy16(const void* g, void* l) {
#if __has_builtin(__builtin_amdgcn_global_load_async_to_lds_b128)
  __builtin_amdgcn_global_load_async_to_lds_b128(
      (__attribute__((address_space(1))) v4i_t*)g,
      (__attribute__((address_space(3))) v4i_t*)l,
      /*imm offset*/ 0, /*cpol*/ 0);
#else
  *(float4*)l = *(const float4*)g;   // fallback: register-staged copy
#endif
}

__device__ __forceinline__ void async_wait_all() {
#if __has_builtin(__builtin_amdgcn_s_wait_asynccnt)
  __builtin_amdgcn_s_wait_asynccnt(0);   // s_wait_asynccnt 0
#endif
}

// ---- kernel 1: init bbox sentinels ---------------------------------------
// ws layout per batch (8 ints): [0]=p_ymin [1]=p_xmin [2]=p_ymax [3]=p_xmax
//                               [4]=t_ymin [5]=t_xmin [6]=t_ymax [7]=t_xmax
__global__ void bbox_init_kernel(int* __restrict__ ws, int n) {
  int i = blockIdx.x * blockDim.x + threadIdx.x;
  if (i < n) {
    int r = i & 7;
    ws[i] = (r == 0 || r == 1 || r == 4 || r == 5) ? INT_MAX : -1;
  }
}

// ---- kernel 2: streaming argmax-mask + bbox reduction ---------------------
__global__ void bbox_scan_kernel(const float* __restrict__ pred,
                                 const float* __restrict__ truth,
                                 int* __restrict__ ws) {
  extern __shared__ __align__(16) unsigned char smem[];
  float* s = (float*)smem;
  const int tid = threadIdx.x;
  const long long tileBase = (long long)blockIdx.x * TILE_PIX;   // global pixel index
  const int batch = (int)(tileBase >> 18);                       // / (512*512)
  const int inBatchBase = (int)(tileBase & 262143LL);
  int* wsb = ws + batch * 8;

  const float* srcs[2] = {pred, truth};

  for (int phase = 0; phase < 2; ++phase) {
    const unsigned char* gbase =
        (const unsigned char*)srcs[phase] + (size_t)tileBase * (CH * 4);

    // Stage 43008 contiguous bytes -> LDS: 21 rounds of wave-wide b128 async DMA
#pragma unroll
    for (int r = 0; r < COPY_ROUNDS; ++r) {
      const int off = (r * BLOCK + tid) * 16;
      async_copy16(gbase + off, smem + off);
    }
    async_wait_all();
    __syncthreads();

    // Per-thread mask + bbox over 4 pixels; LDS reads stride 21 words -> bank-conflict-free
    int ymin = INT_MAX, xmin = INT_MAX, ymax = -1, xmax = -1;
#pragma unroll
    for (int k = 0; k < PIX_PER_THREAD; ++k) {
      const int p = tid + k * BLOCK;
      const float* q = s + p * CH;
      const float p0 = q[0];
      float m = q[1];
#pragma unroll
      for (int c = 2; c < CH; ++c) m = fmaxf(m, q[c]);
      if (m > p0) {                      // strict > : matches jnp.argmax tie-to-0
        const int gi = inBatchBase + p;
        const int y = gi >> 9;           // W = 512
        const int x = gi & 511;
        ymin = min(ymin, y); ymax = max(ymax, y);
        xmin = min(xmin, x); xmax = max(xmax, x);
      }
    }

    // wave32 tree reduction
#pragma unroll
    for (int off = 16; off > 0; off >>= 1) {
      ymin = min(ymin, __shfl_xor(ymin, off));
      xmin = min(xmin, __shfl_xor(xmin, off));
      ymax = max(ymax, __shfl_xor(ymax, off));
      xmax = max(xmax, __shfl_xor(xmax, off));
    }
    if ((tid & 31) == 0 && ymax >= 0) {
      atomicMin(&wsb[phase * 4 + 0], ymin);
      atomicMin(&wsb[phase * 4 + 1], xmin);
      atomicMax(&wsb[phase * 4 + 2], ymax);
      atomicMax(&wsb[phase * 4 + 3], xmax);
    }
    __syncthreads();   // LDS reused by next phase
  }
}

// ---- kernel 3: per-batch penalty + mean -----------------------------------
__global__ void bbox_final_kernel(const int* __restrict__ ws,
                                  float* __restrict__ out, int nB) {
  const int b = threadIdx.x;
  float pen = 0.0f;
  if (b < nB) {
    const int* w = ws + b * 8;
    const bool hp = (w[2] >= 0);
    const bool ht = (w[6] >= 0);
    float py0, px0, py1, px1, ty0, tx0, ty1, tx1;
    if (hp) { py0 = (float)w[0]; px0 = (float)w[1]; py1 = (float)w[2]; px1 = (float)w[3]; }
    else    { py0 = 0.f; px0 = 0.f; py1 = 1.f; px1 = 1.f; }
    if (ht) { ty0 = (float)w[4]; tx0 = (float)w[5]; ty1 = (float)w[6]; tx1 = (float)w[7]; }
    else    { ty0 = 0.f; tx0 = 0.f; ty1 = 1.f; tx1 = 1.f; }

    const float pa = (py1 - py0 + 1.f) * (px1 - px0 + 1.f);
    const float ta = (ty1 - ty0 + 1.f) * (tx1 - tx0 + 1.f);
    const float area_pen = fmaxf(pa - ta, 0.f) / (ta + 1.f);

    const float dy = (py0 + py1) * 0.5f - (ty0 + ty1) * 0.5f;
    const float dx = (px0 + px1) * 0.5f - (tx0 + tx1) * 0.5f;
    const float center = sqrtf(dy * dy + dx * dx) * (1.0f / 20.0f);

    const float iy0 = fmaxf(py0, ty0), ix0 = fmaxf(px0, tx0);
    const float iy1 = fminf(py1, ty1), ix1 = fminf(px1, tx1);
    const float ia = fmaxf(0.f, iy1 - iy0 + 1.f) * fmaxf(0.f, ix1 - ix0 + 1.f);
    const float ua = pa + ta - ia + 1e-6f;
    const float iou_pen = 1.f - ia / ua;

    pen = (hp && ht) ? tanhf(area_pen + center + iou_pen) : 0.f;
  }
#pragma unroll
  for (int off = 16; off > 0; off >>= 1) pen += __shfl_xor(pen, off);
  if (b == 0) out[0] = 0.05f * pen / (float)nB;
}

// ---- launcher -------------------------------------------------------------
extern "C" void kernel_launch(void* const* d_in, const int* in_sizes, int n_in,
                              void* d_out, int out_size, void* d_ws, size_t ws_size,
                              hipStream_t stream) {
  const float* pred  = (const float*)d_in[0];
  const float* truth = (const float*)d_in[1];
  float* out = (float*)d_out;
  int* ws = (int*)d_ws;

  const int HWC = 512 * 512 * CH;
  const int nB = in_sizes[0] / HWC;          // = 32 for the reference shapes

  bbox_init_kernel<<<1, 256, 0, stream>>>(ws, nB * 8);

  const int tiles = nB * (512 * 512) / TILE_PIX;   // 16384 blocks
  bbox_scan_kernel<<<tiles, BLOCK, TILE_BYTES, stream>>>(pred, truth, ws);

  bbox_final_kernel<<<1, 32, 0, stream>>>(ws, out, nB);
}